// AttLayer_6528350290211
// MI455X (gfx1250) — compile-verified
//
#include <hip/hip_runtime.h>
#include <hip/hip_bf16.h>
#include <stdint.h>

#define NUM_GRAPHS 4096
#define EMB 128
#define NW 6                         /* waves per block (wave32) */
#define BLOCK (NW * 32)
#define TILE_ROWS 16
#define TILE_FLOATS (TILE_ROWS * EMB) /* 2048 floats = 8KB per wave tile */

typedef float v2f __attribute__((ext_vector_type(2)));
typedef float v8f __attribute__((ext_vector_type(8)));

// Generic (flat) shared pointer -> 32-bit workgroup-relative LDS byte offset,
// as required by the VDST operand of global_load_async_to_lds_*.
__device__ __forceinline__ uint32_t lds_offset(const void* p) {
  return (uint32_t)(uintptr_t)(void __attribute__((address_space(3))) *)(void*)(uintptr_t)p;
}

__global__ __launch_bounds__(BLOCK) void att_pool_kernel(
    const float* __restrict__ x, const int* __restrict__ batch,
    const float* __restrict__ w, float* __restrict__ out, int n_nodes) {
  // Single shared block: NW per-wave tiles (reused as per-wave combine area) + staged w.
  __shared__ float smem[NW * TILE_FLOATS + EMB];
  float* wlds = &smem[NW * TILE_FLOATS];

  const int g    = blockIdx.x;
  const int tid  = threadIdx.x;
  const int lane = tid & 31;
  const int wv   = tid >> 5;

  // Segment bounds: batch is sorted; redundant uniform binary search (broadcast loads).
  int lo = 0, hi = n_nodes;
  while (lo < hi) { int mid = (lo + hi) >> 1; if (batch[mid] < g) lo = mid + 1; else hi = mid; }
  const int seg_begin = lo;
  hi = n_nodes;
  while (lo < hi) { int mid = (lo + hi) >> 1; if (batch[mid] < g + 1) lo = mid + 1; else hi = mid; }
  const int seg_end = lo;
  const int cnt = seg_end - seg_begin;

  // Pass-through output: att_weight copy (block 0 only).
  if (g == 0 && tid < EMB) out[NUM_GRAPHS * EMB + tid] = w[tid];

  // Empty segment -> zeros (matches reference: counts clamped to 1, weighted_sum 0).
  if (cnt == 0) { if (tid < EMB) out[g * EMB + tid] = 0.0f; return; }

  if (tid < EMB) wlds[tid] = w[tid];
  __syncthreads();

  float* tile = &smem[wv * TILE_FLOATS];
  const char* tbytes = (const char*)tile;
  const uint32_t tile_lds = lds_offset(tile);

  // Per-wave online softmax state: m=running max, s=running denom,
  // acc = running weighted sum; lane holds dims [4*lane .. 4*lane+3].
  float m = -1e30f, s = 0.0f;
  float acc0 = 0.f, acc1 = 0.f, acc2 = 0.f, acc3 = 0.f;

  const int r  = lane & 15;  // WMMA A-matrix row owned by this lane
  const int hh = lane >> 4;  // lane half selects K sub-pair (A layout per ISA 7.12.2)

  const int ntiles = (cnt + TILE_ROWS - 1) / TILE_ROWS;
  for (int t = wv; t < ntiles; t += NW) {
    const int node0 = seg_begin + t * TILE_ROWS;
    const int tcnt  = (seg_end - node0 < TILE_ROWS) ? (seg_end - node0) : TILE_ROWS;

    // Ensure last iteration's LDS reads retired before async engine overwrites the tile.
    asm volatile("s_wait_dscnt 0" ::: "memory");

    // Async bulk copy global->LDS: one instruction per 512B row (32 lanes x 16B).
    // LDS destination is XOR-swizzled per row (16B group ^= row) so later
    // ds_read_b64 A-gathers and ds_read_b128 row reads are bank-conflict-free.
    const uint64_t grow0 =
        (uint64_t)(uintptr_t)x + ((uint64_t)node0 * EMB + (uint32_t)lane * 4u) * 4u;
#pragma unroll
    for (int rr = 0; rr < TILE_ROWS; ++rr) {
      const uint32_t ldst = tile_lds + (uint32_t)rr * 512u + (uint32_t)((lane ^ rr) << 4);
      if (rr < tcnt) {
        const uint64_t ga = grow0 + (uint64_t)rr * 512u;
        asm volatile("global_load_async_to_lds_b128 %0, %1, off"
                     :: "v"(ldst), "v"(ga) : "memory");
      } else {
        // Zero-fill padded rows: keeps WMMA inputs finite; padded scores (=0)
        // only affect the running max, which cancels in the softmax algebra.
        *(float4*)((char*)tile + rr * 512 + ((lane ^ rr) << 4)) = make_float4(0.f, 0.f, 0.f, 0.f);
      }
    }
    asm volatile("s_wait_asynccnt 0" ::: "memory");

    // Scores for 16 nodes as one chained f32 WMMA GEMM:
    //   D(16x16) += A(16x4: node rows, K=dim chunk) * B(4x16: w chunk broadcast over N).
    // A layout (ISA 7.12.2): lanes 0-15 hold K=2*half+{0,1} of row M=lane&15.
    // B mirrors A's K mapping; identical values across N => every D column = scores.
    v8f c = {};
#pragma unroll
    for (int ck = 0; ck < 32; ++ck) {
      v2f a = *(const v2f*)(tbytes + r * 512 + ((ck ^ r) << 4) + hh * 8);
      v2f b = *(const v2f*)((const char*)wlds + ck * 16 + hh * 8);
      c = __builtin_amdgcn_wmma_f32_16x16x4_f32(false, a, false, b, (short)0, c,
                                                false, false);
    }
    float carr[8];
#pragma unroll
    for (int i = 0; i < 8; ++i) carr[i] = c[i];

    // Tile max (D layout: VGPR i / lane-half h -> node i + 8h), then online rescale.
    float tmax = carr[0];
#pragma unroll
    for (int i = 1; i < 8; ++i) tmax = fmaxf(tmax, carr[i]);
    tmax = fmaxf(tmax, __shfl_xor(tmax, 16, 32));
    const float newm  = fmaxf(m, tmax);
    const float scale = __expf(m - newm);
    s *= scale; acc0 *= scale; acc1 *= scale; acc2 *= scale; acc3 *= scale;
    m = newm;

    // Weighted accumulate: conflict-free ds_read_b128 per node row.
#pragma unroll
    for (int i = 0; i < TILE_ROWS; ++i) {
      if (i < tcnt) {
        const float sc = __shfl(carr[i & 7], (i >> 3) << 4, 32);
        const float e  = __expf(sc - m);
        s += e;
        const float4 xv = *(const float4*)(tbytes + i * 512 + ((lane ^ i) << 4));
        acc0 += e * xv.x; acc1 += e * xv.y; acc2 += e * xv.z; acc3 += e * xv.w;
      }
    }
  }

  // Publish per-wave partial softmax state into this wave's tile region.
  asm volatile("s_wait_asynccnt 0" ::: "memory");
  if (lane == 0) { tile[0] = m; tile[1] = s; }
  *(float4*)(tile + 8 + 4 * lane) = make_float4(acc0, acc1, acc2, acc3);
  __syncthreads();

  // Merge NW wave states: out[d] = (sum_w exp(m_w-M)*acc_w[d]) / (sum_w s_w*exp(m_w-M)) / cnt
  if (tid < EMB) {
    float M = -1e30f;
#pragma unroll
    for (int j = 0; j < NW; ++j) M = fmaxf(M, smem[j * TILE_FLOATS + 0]);
    float denom = 0.f, val = 0.f;
#pragma unroll
    for (int j = 0; j < NW; ++j) {
      const float f = __expf(smem[j * TILE_FLOATS + 0] - M);
      denom += smem[j * TILE_FLOATS + 1] * f;
      val   += f * smem[j * TILE_FLOATS + 8 + tid];
    }
    out[g * EMB + tid] = (val / fmaxf(denom, 1e-30f)) / (float)cnt;
  }
}

extern "C" void kernel_launch(void* const* d_in, const int* in_sizes, int n_in,
                              void* d_out, int out_size, void* d_ws, size_t ws_size,
                              hipStream_t stream) {
  (void)n_in; (void)out_size; (void)d_ws; (void)ws_size;
  const float* x     = (const float*)d_in[0];
  const int*   batch = (const int*)d_in[1];
  const float* w     = (const float*)d_in[2];
  float*       out   = (float*)d_out;
  const int n_nodes  = in_sizes[1];
  att_pool_kernel<<<dim3(NUM_GRAPHS), dim3(BLOCK), 0, stream>>>(x, batch, w, out, n_nodes);
}